// SparseEdgeEmbeddingV3_61048665145758
// MI455X (gfx1250) — compile-verified
//
#include <hip/hip_runtime.h>

// CDNA5 / gfx1250: wave32, WMMA f32 16x16x4 computes the pairwise-d^2 tile.
// Bandwidth-bound problem: 604 MB of f32 output @ 23.3 TB/s => ~26 us floor.
// Strategy: 1 v_wmma_f32_16x16x4_f32 per 16x16 tile of d^2 (exact f32, matches
// reference precision), exp2-based mask + per-channel Gaussians, 128-bit
// non-temporal stores (output >> 192MB L2). v_rcp_f32 instead of IEEE div for
// the 16 uniform exponent coefficients.

typedef __attribute__((ext_vector_type(2))) float v2f;
typedef __attribute__((ext_vector_type(4))) float v4f;
typedef __attribute__((ext_vector_type(8))) float v8f;

#define NPTS 3072
#define NCH  16
#define TILE 16
#define WAVES_PER_BLOCK 8

__global__ __launch_bounds__(256) void
sparse_edge_embed_wmma(const float* __restrict__ coord,
                       const float* __restrict__ sigma,
                       float* __restrict__ out) {
    const int lane = threadIdx.x & 31;
    const int wave = threadIdx.x >> 5;
    const int half = lane >> 4;        // 0: lanes 0-15, 1: lanes 16-31
    const int lm   = lane & 15;

    const int i0 = blockIdx.y * TILE;                            // tile rows
    const int j0 = (blockIdx.x * WAVES_PER_BLOCK + wave) * TILE; // tile cols

    // Per-lane source points.
    const int ia = i0 + lm;   // A-matrix row point (duplicated in both halves)
    const int jb = j0 + lm;   // B-matrix col point (duplicated in both halves)

    const float ax = coord[ia * 3 + 0];
    const float ay = coord[ia * 3 + 1];
    const float az = coord[ia * 3 + 2];
    const float bx = coord[jb * 3 + 0];
    const float by = coord[jb * 3 + 1];
    const float bz = coord[jb * 3 + 2];
    const float sqa = ax * ax + ay * ay + az * az;   // |x_i|^2
    const float sqb = bx * bx + by * by + bz * bz;   // |x_j|^2

    // A (16x4 f32, MxK): lane m holds K=0,1 (half 0) or K=2,3 (half 1).
    //   A[m] = [x, y, z, 1]
    // B (4x16 f32, KxN): lane n holds K=0,1 (half 0) or K=2,3 (half 1).
    //   B[:,n] = [-2x, -2y, -2z, |x_j|^2]
    // => (A*B)[m][n] = -2*dot(x_i,x_j) + |x_j|^2 ; add C[m][n] = |x_i|^2.
    v2f a = half ? (v2f){az, 1.0f}        : (v2f){ax, ay};
    v2f b = half ? (v2f){-2.0f * bz, sqb} : (v2f){-2.0f * bx, -2.0f * by};

    // C/D layout: VGPR r, lanes 0-15 -> row M=r, lanes 16-31 -> row M=r+8.
    // Lane l (either half) holds sq of point i0 + (l%16): shuffle to rows.
    v8f c;
#pragma unroll
    for (int r = 0; r < 8; ++r)
        c[r] = __shfl(sqa, r + half * 8);

    // D = A*B + C : full 16x16 tile of squared pairwise distances.
    v8f d = __builtin_amdgcn_wmma_f32_16x16x4_f32(
        /*neg_a=*/false, a, /*neg_b=*/false, b,
        /*c_mod=*/(short)0, c, /*reuse_a=*/false, /*reuse_b=*/false);

    // Per-channel exponent coefficients: k_c(d2) = exp2(d2 * n2[c]),
    // n2[c] = -log2(e) / (2*sigma_c^2). v_rcp_f32 (1 ulp) instead of the
    // ~10-instruction IEEE divide sequence — feeds an exp argument, so the
    // approximation is far below the exp's own error budget.
    const float LOG2E = 1.4426950408889634f;
    float n2[NCH];
#pragma unroll
    for (int ch = 0; ch < NCH; ++ch) {
        const float s = sigma[ch];
        n2[ch] = -LOG2E * __builtin_amdgcn_rcpf(2.0f * s * s);
    }

    const int j = j0 + lm;
#pragma unroll
    for (int r = 0; r < 8; ++r) {
        const int irow = i0 + r + half * 8;
        const float d2 = fmaxf(d[r], 0.0f);

        // Channel 15 uses sigma_max: its value IS the mask test quantity.
        const float k15 = __builtin_amdgcn_exp2f(d2 * n2[15]);
        const bool keep = (k15 >= 0.9f) && (d2 > 0.0f) && (irow != j);

        float vals[NCH];
        if (keep) {
#pragma unroll
            for (int ch = 0; ch < 15; ++ch)
                vals[ch] = __builtin_amdgcn_exp2f(d2 * n2[ch]);
            vals[15] = k15;
        } else {
            // Most tiles land here with EXEC=0 on the branch above.
#pragma unroll
            for (int ch = 0; ch < NCH; ++ch) vals[ch] = 0.0f;
        }

        // out[irow][j][0..15]: 64B contiguous per (i,j); lanes 0-15 cover
        // consecutive j -> coalesced 1KB runs. NT: bypass L2 retention.
        const unsigned long long base =
            ((unsigned long long)irow * NPTS + (unsigned long long)j) * NCH;
        v4f* p = (v4f*)(out + base);
#pragma unroll
        for (int q = 0; q < 4; ++q) {
            v4f v4 = {vals[4 * q + 0], vals[4 * q + 1],
                      vals[4 * q + 2], vals[4 * q + 3]};
            __builtin_nontemporal_store(v4, p + q);
        }
    }
}

extern "C" void kernel_launch(void* const* d_in, const int* in_sizes, int n_in,
                              void* d_out, int out_size, void* d_ws, size_t ws_size,
                              hipStream_t stream) {
    (void)in_sizes; (void)n_in; (void)out_size; (void)d_ws; (void)ws_size;
    const float* coord = (const float*)d_in[0];   // [3072, 3] f32
    const float* sigma = (const float*)d_in[1];   // [16] f32
    float* out = (float*)d_out;                   // [1, 3072, 3072, 16] f32

    // 192 i-tiles x 24 blocks; each block = 8 waves = 8 j-tiles of 16.
    dim3 grid(NPTS / (TILE * WAVES_PER_BLOCK), NPTS / TILE, 1);
    sparse_edge_embed_wmma<<<grid, 32 * WAVES_PER_BLOCK, 0, stream>>>(coord, sigma, out);
}